// BilateralBlur_65292092834210
// MI455X (gfx1250) — compile-verified
//
#include <hip/hip_runtime.h>
#include <hip/hip_bf16.h>
#include <stdint.h>
#include <math.h>

// ---------------------------------------------------------------------------
// Bilateral 7x7 blur, (B=4, C=3, H=512, W=512) fp32, reflect padding.
// CDNA5 strategy: async global->LDS tile staging (ASYNCcnt path) + LDS-tiled
// 7x7 stencil with cross-channel color weights. Wave32, 8 waves / block.
// ---------------------------------------------------------------------------

#define KR       3
#define KW       7
#define TILE_W   32
#define TILE_H   8
#define LDS_W    (TILE_W + 2 * KR)   // 38
#define LDS_P    40                  // padded row stride (floats)
#define LDS_H    (TILE_H + 2 * KR)   // 14
#define IMG_H    512
#define IMG_W    512
#define NCH      3
#define NTHREADS 256

#if defined(__has_builtin)
#if __has_builtin(__builtin_amdgcn_global_load_async_to_lds_b32)
#define HAVE_ASYNC_LDS_BUILTIN 1
#endif
#if __has_builtin(__builtin_amdgcn_s_wait_asynccnt)
#define HAVE_WAIT_ASYNC_BUILTIN 1
#endif
#if __has_builtin(__builtin_amdgcn_exp2f)
#define HAVE_EXP2_BUILTIN 1
#endif
#endif

typedef __attribute__((address_space(1))) int GInt;   // global int
typedef __attribute__((address_space(3))) int LInt;   // LDS int

__device__ __forceinline__ int reflect_idx(int i, int n) {
  // jnp.pad 'reflect' (no edge repeat); pad <= 3 so one application suffices
  i = (i < 0) ? -i : i;
  return (i >= n) ? (2 * n - 2 - i) : i;
}

__device__ __forceinline__ void async_g2l_b32(const float* g, float* l) {
#if defined(HAVE_ASYNC_LDS_BUILTIN)
  // global_load_async_to_lds_b32 : copies 4 bytes from the global address g to the LDS address l; tracked by ASYNCcnt
  __builtin_amdgcn_global_load_async_to_lds_b32(
      (GInt*)(uintptr_t)g, (LInt*)(uintptr_t)l, /*offset=*/0, /*cpol=*/0);
#else
  unsigned ldsa = (unsigned)(uintptr_t)l;
  asm volatile("global_load_async_to_lds_b32 %0, %1, off"
               :: "v"(ldsa), "v"(g)
               : "memory");
#endif
}

__device__ __forceinline__ void wait_async0() {
#if defined(HAVE_WAIT_ASYNC_BUILTIN)
  __builtin_amdgcn_s_wait_asynccnt(0);
#else
  asm volatile("s_wait_asynccnt 0" ::: "memory");
#endif
}

__device__ __forceinline__ float fast_exp2(float x) {
#if defined(HAVE_EXP2_BUILTIN)
  return __builtin_amdgcn_exp2f(x);   // v_exp_f32, no extra scaling mul
#else
  return exp2f(x);                    // host-parseable fallback
#endif
}

__global__ __launch_bounds__(NTHREADS) void
BilateralBlur_65292092834210_kernel(const float* __restrict__ in,
                                    float* __restrict__ out) {
  __shared__ float smem[NCH * LDS_H * LDS_P];   // 6720 bytes

  const int tid = threadIdx.x;
  const int tx  = tid & (TILE_W - 1);
  const int ty  = tid >> 5;                     // 0..7
  const int bx0 = blockIdx.x * TILE_W;
  const int by0 = blockIdx.y * TILE_H;
  const int b   = blockIdx.z;

  // ---- Stage padded tile (3 channels) into LDS via async copies ----------
  const int PLANE = LDS_H * LDS_W;              // 532
  const int TOT   = NCH * PLANE;                // 1596 (~6.2 per thread)
  for (int i = tid; i < TOT; i += NTHREADS) {
    const int c  = i / PLANE;
    const int r  = i - c * PLANE;
    const int ly = r / LDS_W;
    const int lx = r - ly * LDS_W;
    const int gy = reflect_idx(by0 - KR + ly, IMG_H);
    const int gx = reflect_idx(bx0 - KR + lx, IMG_W);
    const float* g = in + (((size_t)(b * NCH + c) * IMG_H) + gy) * IMG_W + gx;
    float* l = &smem[c * (LDS_H * LDS_P) + ly * LDS_P + lx];
    async_g2l_b32(g, l);
  }
  wait_async0();       // my async LDS writes are visible
  __syncthreads();     // everyone's are

  // ---- 7x7 bilateral stencil out of LDS ----------------------------------
  // Normalized 1D Gaussian, sigma = 1.5 (folded to immediates by unroll)
  const float G7[KW] = {0.0366328f, 0.1112757f, 0.2167453f, 0.2706820f,
                        0.2167453f, 0.1112757f, 0.0366328f};
  // exp(-50*d) == exp2(-50*log2(e)*d): fold log2(e) into the constant so the
  // color weight is 2 muls + v_exp_f32 per tap (no extra scaling multiply).
  const float NEG_50_LOG2E = -72.134752f;
#define S(c, y, x) smem[(c) * (LDS_H * LDS_P) + (y) * LDS_P + (x)]

  const float c0 = S(0, ty + KR, tx + KR);
  const float c1 = S(1, ty + KR, tx + KR);
  const float c2 = S(2, ty + KR, tx + KR);

  float acc0 = 0.f, acc1 = 0.f, acc2 = 0.f, wsum = 0.f;
#pragma unroll
  for (int dy = 0; dy < KW; ++dy) {
    const float gy = G7[dy];
#pragma unroll
    for (int dx = 0; dx < KW; ++dx) {
      const float sw = gy * G7[dx];
      const float t0 = S(0, ty + dy, tx + dx);
      const float t1 = S(1, ty + dy, tx + dx);
      const float t2 = S(2, ty + dy, tx + dx);
      const float cd = fabsf(t0 - c0) + fabsf(t1 - c1) + fabsf(t2 - c2);
      const float w  = sw * fast_exp2(NEG_50_LOG2E * cd * cd);
      wsum += w;
      acc0 += w * t0;
      acc1 += w * t1;
      acc2 += w * t2;
    }
  }
#undef S

  const float  inv   = 1.0f / wsum;
  const size_t plane = (size_t)IMG_H * IMG_W;
  const size_t obase =
      ((size_t)(b * NCH) * IMG_H + (size_t)(by0 + ty)) * IMG_W + (bx0 + tx);
  out[obase]             = acc0 * inv;
  out[obase + plane]     = acc1 * inv;
  out[obase + 2 * plane] = acc2 * inv;
}

extern "C" void kernel_launch(void* const* d_in, const int* in_sizes, int n_in,
                              void* d_out, int out_size, void* d_ws, size_t ws_size,
                              hipStream_t stream) {
  const float* in  = (const float*)d_in[0];
  float*       out = (float*)d_out;
  const int B = in_sizes[0] / (NCH * IMG_H * IMG_W);   // 4 for the ref shapes
  dim3 grid(IMG_W / TILE_W, IMG_H / TILE_H, B);        // (16, 64, 4)
  BilateralBlur_65292092834210_kernel<<<grid, dim3(NTHREADS), 0, stream>>>(in, out);
}